// Nonbonded_56513179681294
// MI455X (gfx1250) — compile-verified
//
#include <hip/hip_runtime.h>

// Nonbonded pair energy (LJ + Coulomb, min-image PBC) -> scalar sum.
// HBM-bound on the 64MB pair-index stream; atom tables (~2.4MB) live in L2/WGP$.
// Wave reduction uses V_WMMA_F32_16X16X4_F32 (ones-vector row-sum trick) + ds_swizzle.

typedef __attribute__((ext_vector_type(4))) int   v4i;
typedef __attribute__((ext_vector_type(2))) float v2f;
typedef __attribute__((ext_vector_type(8))) float v8f;

#define BLOCK 256
#define MAX_BLOCKS 16384

// Full wave32 sum via the matrix unit:
// A(16x4): VGPR0 = per-lane partial (lanes0-15 -> A[m][0], lanes16-31 -> A[m][2]), VGPR1 = 0.
// B(4x16) = all ones  =>  D[m][n] = partial[m] + partial[m+16]  (independent of n).
// Sum the 8 D VGPRs per lane (half-sums per 16-lane group), then one SWAPX16 swizzle add.
__device__ __forceinline__ float wave_reduce32(float v) {
    v2f a; a.x = v;    a.y = 0.0f;
    v2f b; b.x = 1.0f; b.y = 1.0f;
    v8f c = {};
    v8f d = __builtin_amdgcn_wmma_f32_16x16x4_f32(
        /*neg_a=*/false, a, /*neg_b=*/false, b,
        /*c_mod=*/(short)0, c, /*reuse_a=*/false, /*reuse_b=*/false);
    float s = d[0] + d[1] + d[2] + d[3] + d[4] + d[5] + d[6] + d[7];
    // SWAPX16: group-of-32 swizzle, xor=0x10, and=0x1f -> offset 0x401F
    float o = __int_as_float(__builtin_amdgcn_ds_swizzle(__float_as_int(s), 0x401F));
    return s + o;   // every lane now holds the full 32-lane sum
}

__device__ __forceinline__ float block_reduce(float v) {
    __shared__ float wsum[BLOCK / 32];
    float w = wave_reduce32(v);
    int lane = threadIdx.x & 31;
    int wid  = threadIdx.x >> 5;
    if (lane == 0) wsum[wid] = w;
    __syncthreads();
    float tot = 0.0f;
    if (threadIdx.x == 0) {
#pragma unroll
        for (int k = 0; k < BLOCK / 32; ++k) tot += wsum[k];
    }
    return tot; // valid on thread 0 only
}

__global__ void __launch_bounds__(BLOCK)
nb_pair_energy_kernel(const float* __restrict__ coords,
                      const int*   __restrict__ pairs,
                      const float* __restrict__ box,
                      const float* __restrict__ sigma,
                      const float* __restrict__ epsil,
                      const float* __restrict__ chg,
                      const float* __restrict__ coulKp,
                      const float* __restrict__ cutoffp,
                      const float* __restrict__ doshiftp,
                      float*       __restrict__ partials,
                      long long n_pairs)
{
    // ---- uniform parameters (lower to scalar loads / SALU) ----
    float b00 = box[0], b01 = box[1], b02 = box[2];
    float b10 = box[3], b11 = box[4], b12 = box[5];
    float b20 = box[6], b21 = box[7], b22 = box[8];
    float det = b00 * (b11 * b22 - b12 * b21)
              - b01 * (b10 * b22 - b12 * b20)
              + b02 * (b10 * b21 - b11 * b20);
    float rdet = 1.0f / det;
    float i00 = (b11 * b22 - b12 * b21) * rdet, i01 = (b02 * b21 - b01 * b22) * rdet, i02 = (b01 * b12 - b02 * b11) * rdet;
    float i10 = (b12 * b20 - b10 * b22) * rdet, i11 = (b00 * b22 - b02 * b20) * rdet, i12 = (b02 * b10 - b00 * b12) * rdet;
    float i20 = (b10 * b21 - b11 * b20) * rdet, i21 = (b01 * b20 - b00 * b21) * rdet, i22 = (b00 * b11 - b01 * b10) * rdet;
    float ck    = coulKp[0];
    float cut   = cutoffp[0];
    float cut2  = cut * cut;
    float shift = doshiftp[0] / cut;

    auto epair = [&](int i, int j) -> float {
        float xi = coords[3 * i], yi = coords[3 * i + 1], zi = coords[3 * i + 2];
        float xj = coords[3 * j], yj = coords[3 * j + 1], zj = coords[3 * j + 2];
        float dx = xi - xj, dy = yi - yj, dz = zi - zj;
        // ds = dr @ box_inv
        float sx = dx * i00 + dy * i10 + dz * i20;
        float sy = dx * i01 + dy * i11 + dz * i21;
        float sz = dx * i02 + dy * i12 + dz * i22;
        sx -= floorf(sx + 0.5f);
        sy -= floorf(sy + 0.5f);
        sz -= floorf(sz + 0.5f);
        // dr_pbc = ds @ box
        float px = sx * b00 + sy * b10 + sz * b20;
        float py = sx * b01 + sy * b11 + sz * b21;
        float pz = sx * b02 + sy * b12 + sz * b22;
        float r2   = px * px + py * py + pz * pz;
        float rinv = __frsqrt_rn(r2);
        float e    = chg[i] * chg[j] * (rinv - shift) * ck;
        float sij  = 0.5f * (sigma[i] + sigma[j]);
        float eij  = sqrtf(epsil[i] * epsil[j]);
        float u  = sij * rinv;
        float u2 = u * u;
        float u6 = u2 * u2 * u2;
        e += 4.0f * eij * u6 * (u6 - 1.0f);
        return (r2 <= cut2) ? e : 0.0f;
    };

    long long nchunks = (n_pairs + 1) >> 1;           // 2 pairs per 128-bit load
    long long tid     = (long long)blockIdx.x * blockDim.x + threadIdx.x;
    long long stride  = (long long)gridDim.x * blockDim.x;
    const v4i* pairs4 = (const v4i*)pairs;

    float acc = 0.0f;
    for (long long c = tid; c < nchunks; c += stride) {
        // prefetch next chunk of the read-once index stream (speculative, OOB-safe)
        __builtin_prefetch(pairs + 4 * (c + stride), 0, 1);
        if (2 * c + 1 < n_pairs) {
            v4i pk = __builtin_nontemporal_load(pairs4 + c); // {i0,j0,i1,j1}, TH=NT
            acc += epair(pk.x, pk.y);
            acc += epair(pk.z, pk.w);
        } else {                                             // odd tail pair
            acc += epair(pairs[4 * c], pairs[4 * c + 1]);
        }
    }

    float tot = block_reduce(acc);
    if (threadIdx.x == 0) partials[blockIdx.x] = tot;
}

__global__ void __launch_bounds__(BLOCK)
nb_reduce_kernel(const float* __restrict__ partials, int n, float* __restrict__ out)
{
    float acc = 0.0f;
    for (int i = threadIdx.x; i < n; i += BLOCK) acc += partials[i]; // fixed order
    float tot = block_reduce(acc);
    if (threadIdx.x == 0) out[0] = tot;
}

extern "C" void kernel_launch(void* const* d_in, const int* in_sizes, int n_in,
                              void* d_out, int out_size, void* d_ws, size_t ws_size,
                              hipStream_t stream)
{
    const float* coords  = (const float*)d_in[0];
    const int*   pairs   = (const int*)  d_in[1];
    const float* box     = (const float*)d_in[2];
    const float* sigma   = (const float*)d_in[3];
    const float* epsil   = (const float*)d_in[4];
    const float* chg     = (const float*)d_in[5];
    const float* coulK   = (const float*)d_in[6];
    const float* cutoff  = (const float*)d_in[7];
    const float* doshift = (const float*)d_in[8];

    long long n_pairs = (long long)in_sizes[1] / 2;
    long long nchunks = (n_pairs + 1) / 2;
    long long want    = (nchunks + BLOCK - 1) / BLOCK;
    long long cap     = (long long)(ws_size / sizeof(float));
    long long nb      = want < (long long)MAX_BLOCKS ? want : (long long)MAX_BLOCKS;
    if (nb > cap) nb = cap;
    if (nb < 1)   nb = 1;

    float* partials = (float*)d_ws;

    nb_pair_energy_kernel<<<(int)nb, BLOCK, 0, stream>>>(
        coords, pairs, box, sigma, epsil, chg, coulK, cutoff, doshift,
        partials, n_pairs);

    nb_reduce_kernel<<<1, BLOCK, 0, stream>>>(partials, (int)nb, (float*)d_out);
}